// FlyVisAdExODE_71786083385561
// MI455X (gfx1250) — compile-verified
//
#include <hip/hip_runtime.h>
#include <stdint.h>

// ---------------------------------------------------------------------------
// AdEx network Euler step, MI455X (gfx1250).
//   Phase 0: fuse spiked/is_excitatory -> 1B code table, zero counters
//   Phase 1: edge scatter; edge_index streamed through LDS via the CDNA5
//            Tensor Data Mover (tensor_load_to_lds, TENSORcnt) for the src
//            half and per-wave global_load_async_to_lds_b128 (ASYNCcnt) for
//            the dst half; one byte gather + one u32 atomic per active edge.
//   Phase 2: per-neuron elementwise ODE, applies Q_* * count.
// Roofline: ~84 MB traffic/step -> ~3.6 us at 23.3 TB/s; FLOPs negligible.
// No dense matmul exists in this op, so the CDNA5 features exercised are the
// async/tensor data movers, not WMMA.
// ---------------------------------------------------------------------------

constexpr int TPB         = 256;      // 8 waves (wave32)
constexpr int TILE_DW     = 4096;     // edges per block tile (16 KB per array)
constexpr int DW_PER_WAVE = TILE_DW / (TPB / 32);   // 512 dwords per wave
constexpr int EDGE_BLOCKS = 1024;

typedef unsigned int u32x4 __attribute__((ext_vector_type(4)));
typedef int          i32x8 __attribute__((ext_vector_type(8)));
typedef int          i32x4 __attribute__((ext_vector_type(4)));

#if __has_builtin(__builtin_amdgcn_tensor_load_to_lds) && \
    __has_builtin(__builtin_amdgcn_s_wait_tensorcnt)
#define HAVE_TDM 1
#else
#define HAVE_TDM 0
#endif

// low 32 bits of a generic pointer to a __shared__ object == LDS byte address
__device__ __forceinline__ uint32_t lds_addr_of(const void* p) {
  return (uint32_t)(uintptr_t)p;
}

// ---- fine-grained CDNA5 async global->LDS copy (per-lane 16B, ASYNCcnt) ----
__device__ __forceinline__ void async_copy_b128(uint32_t lds_byte_addr,
                                                const void* gaddr) {
  asm volatile("global_load_async_to_lds_b128 %0, %1, off"
               :
               : "v"(lds_byte_addr), "v"((uint64_t)(uintptr_t)gaddr)
               : "memory");
}
__device__ __forceinline__ void wait_async_le4() {
  asm volatile("s_wait_asynccnt 4" ::: "memory");
}
__device__ __forceinline__ void wait_async_0() {
  asm volatile("s_wait_asynccnt 0" ::: "memory");
}

#if HAVE_TDM
// ---- TDM: one instruction DMAs a contiguous 1-row tile of n_dw dwords ------
// D# layout per CDNA5 ISA ch.8: group0 = {count|..., lds_addr, global_addr,
// type=2}; group1 = {data_size=4B, tensor_dim0/1, tile_dim0/1, dim0_stride}.
// This toolchain exposes the 6-arg builtin:
//   (u32x4 g0, i32x8 g1, i32x4 g2, i32x4 g3, i32x8 extra, i32 cpol)
__device__ __forceinline__ void tdm_load_1d(uint32_t lds_byte,
                                            const void* gsrc,
                                            uint32_t n_dw) {
  const uint64_t ga  = (uint64_t)(uintptr_t)gsrc;
  const uint32_t TD0 = 0x40000000u;            // huge tensor_dim0: no OOB clip
  u32x4 g0;
  g0[0] = 1u;                                   // count=1 (valid), no gather
  g0[1] = lds_byte;                             // lds_addr (bytes)
  g0[2] = (uint32_t)(ga & 0xFFFFFFFFu);         // global_addr[31:0]
  g0[3] = (uint32_t)((ga >> 32) & 0x01FFFFFFu)  // global_addr[56:32]
        | (2u << 30);                           // type = 2 ("image")
  i32x8 g1;
  g1[0] = (int)(2u << 16);                      // wg_mask=0, data_size=4B
  g1[1] = (int)((TD0 & 0xFFFFu) << 16);         // barrier_addr=0 | dim0.lo16
  g1[2] = (int)(((TD0 >> 16) & 0xFFFFu)         // dim0.hi16
        | (1u << 16));                          // tensor_dim1.lo16 = 1
  g1[3] = (int)(n_dw << 16);                    // dim1.hi16=0 | tile_dim0
  g1[4] = (int)1u;                              // tile_dim1=1, tile_dim2=0
  g1[5] = (int)n_dw;                            // tensor_dim0_stride.lo32
  g1[6] = 0;                                    // stride0.hi16 | stride1.lo16
  g1[7] = 0;
  const i32x4 gz4 = {0, 0, 0, 0};               // groups 2/3 unused (<=2D)
  const i32x8 gz8 = {0, 0, 0, 0, 0, 0, 0, 0};   // extra group unused
  __builtin_amdgcn_tensor_load_to_lds(g0, g1, gz4, gz4, gz8, 0);
}
__device__ __forceinline__ void wait_src_tile(bool more) {
  if (more) __builtin_amdgcn_s_wait_tensorcnt(1);
  else      __builtin_amdgcn_s_wait_tensorcnt(0);
}
#else
// Fallback: wave0 stages the src tile with 32 async b128 copies.
__device__ __forceinline__ void tdm_load_1d(uint32_t lds_byte,
                                            const void* gsrc,
                                            uint32_t n_dw) {
  const int lane = threadIdx.x & 31;
  const int* g = (const int*)gsrc;
  for (uint32_t i = lane * 4; i < n_dw; i += 128)
    async_copy_b128(lds_byte + i * 4u, g + i);
}
__device__ __forceinline__ void wait_src_tile(bool) { wait_async_0(); }
#endif

// ---------------------------------------------------------------------------
// Phase 0: build 1-byte spike code table + zero interleaved counters
//   code[i] = spiked ? (is_exc ? 1 : 2) : 0
// ---------------------------------------------------------------------------
__global__ __launch_bounds__(TPB) void init_kernel(
    const unsigned char* __restrict__ spiked,
    const unsigned char* __restrict__ is_exc,
    unsigned char* __restrict__ code,
    unsigned int* __restrict__ cnt2, int n) {
  int i = blockIdx.x * TPB + threadIdx.x;
  if (i < n) {
    code[i] = spiked[i] ? (is_exc[i] ? 1u : 2u) : 0u;
    cnt2[2 * i + 0] = 0u;
    cnt2[2 * i + 1] = 0u;
  }
}

__device__ __forceinline__ void process_edge(int s, int d,
                                             const unsigned char* __restrict__ code,
                                             unsigned int* __restrict__ cnt2) {
  const unsigned c = code[s];            // single L2-resident byte gather
  if (c)                                 // ~5% of edges
    atomicAdd(&cnt2[2 * d + (c - 1u)], 1u);
}

// ---------------------------------------------------------------------------
// Phase 1: block-tiled edge accumulation.
//   src tile: TDM DMA (wave0 issues, TENSORcnt)      -> shared by block
//   dst tile: per-wave async b128 copies (ASYNCcnt)  -> wave-local slices
// ---------------------------------------------------------------------------
__global__ __launch_bounds__(TPB) void edge_accum_kernel(
    const int* __restrict__ e_src, const int* __restrict__ e_dst, long E,
    const unsigned char* __restrict__ code,
    unsigned int* __restrict__ cnt2) {
  __shared__ __align__(16) int lds_src[2][TILE_DW];
  __shared__ __align__(16) int lds_dst[2][TILE_DW];

  const int tid  = threadIdx.x;
  const int lane = tid & 31;
  const int widx = tid >> 5;

  const long total_tiles = E / TILE_DW;
  const long nb  = gridDim.x;
  const long per = total_tiles / nb;
  const long rem = total_tiles % nb;
  const long bid = blockIdx.x;
  const long t0  = bid * per + (bid < rem ? bid : rem);
  const long t1  = t0 + per + (bid < rem ? 1 : 0);
  const long nt  = t1 - t0;

  auto stage_src = [&](long t, int b) {          // wave0 only
    tdm_load_1d(lds_addr_of(&lds_src[b][0]), e_src + t * (long)TILE_DW,
                (uint32_t)TILE_DW);
  };
  auto stage_dst = [&](long t, int b) {          // every wave, its own slice
#pragma unroll
    for (int i = 0; i < 4; ++i) {
      const int off = widx * DW_PER_WAVE + i * 128 + lane * 4;
      async_copy_b128(lds_addr_of(&lds_dst[b][off]),
                      e_dst + t * (long)TILE_DW + off);
    }
  };

  if (nt > 0) {
    if (widx == 0) stage_src(t0, 0);
    stage_dst(t0, 0);
    if (nt > 1) {
      if (widx == 0) stage_src(t0 + 1, 1);
      stage_dst(t0 + 1, 1);
    }
    int buf = 0;
    for (long t = t0; t < t1; ++t) {
      const bool more = (t + 1 < t1);
      if (widx == 0) wait_src_tile(more);        // TENSORcnt (or fallback)
      if (more) wait_async_le4(); else wait_async_0();   // own dst slice
      __syncthreads();                           // publish buf to the block
#pragma unroll
      for (int k = 0; k < 4; ++k) {
        const int idx = tid * 16 + k * 4;
        const int4 s4 = *(const int4*)&lds_src[buf][idx];
        const int4 d4 = *(const int4*)&lds_dst[buf][idx];
        process_edge(s4.x, d4.x, code, cnt2);
        process_edge(s4.y, d4.y, code, cnt2);
        process_edge(s4.z, d4.z, code, cnt2);
        process_edge(s4.w, d4.w, code, cnt2);
      }
      __syncthreads();                           // all reads of buf done
      if (t + 2 < t1) {
        if (widx == 0) stage_src(t + 2, buf);
        stage_dst(t + 2, buf);
      }
      buf ^= 1;
    }
  }

  // tail edges (E % TILE_DW) handled by the last block with direct loads
  if (bid == nb - 1) {
    for (long e = total_tiles * (long)TILE_DW + tid; e < E; e += TPB)
      process_edge(e_src[e], e_dst[e], code, cnt2);
  }
}

// ---------------------------------------------------------------------------
// Phase 2: per-neuron AdEx Euler step
// ---------------------------------------------------------------------------
__global__ __launch_bounds__(TPB) void neuron_kernel(
    const float* __restrict__ voltage, const float* __restrict__ w_in,
    const float* __restrict__ ge_in,   const float* __restrict__ gi_in,
    const float* __restrict__ stim,    const float* __restrict__ refr_in,
    const float* __restrict__ g_L,     const float* __restrict__ delta_T,
    const float* __restrict__ v_thr,   const float* __restrict__ v_rest,
    const float* __restrict__ Cm,      const float* __restrict__ a_p,
    const float* __restrict__ b_p,     const float* __restrict__ tau_w,
    const float* __restrict__ tau_ge,  const float* __restrict__ tau_gi,
    const float* __restrict__ E_ge,    const float* __restrict__ E_gi,
    const float* __restrict__ I_bias,  const float* __restrict__ stim_scale,
    const float* __restrict__ Q_ge,    const float* __restrict__ Q_gi,
    const float* __restrict__ v_cut,   const float* __restrict__ v_reset,
    const float* __restrict__ t_refrac,
    const unsigned int* __restrict__ cnt2,
    const float* __restrict__ dt_ptr,
    float* __restrict__ out_v,  float* __restrict__ out_w,
    float* __restrict__ out_ge, float* __restrict__ out_gi,
    float* __restrict__ out_rf, float* __restrict__ out_spk,
    int n) {
  const int i = blockIdx.x * TPB + threadIdx.x;
  if (i >= n) return;
  const float dt = dt_ptr[0];

  const float v = voltage[i];
  const float w = w_in[i];
  float ge = ge_in[i] + Q_ge[i] * (float)cnt2[2 * i + 0];  // post-propagation
  float gi = gi_in[i] + Q_gi[i] * (float)cnt2[2 * i + 1];

  const float gl = g_L[i], dT = delta_T[i], vt = v_thr[i], vr = v_rest[i];

  const float I  = I_bias[i] + stim_scale[i] * stim[i]
                 + ge * (E_ge[i] - v) + gi * (E_gi[i] - v);
  const float ex = gl * dT * __expf(fminf((v - vt) / dT, 20.0f));
  const float dv = (-gl * (v - vr) + ex - w + I) / Cm[i];
  const float dw = (-w + a_p[i] * (v - vr)) / tau_w[i];

  const float refr = refr_in[i];
  float vn = (refr <= 0.0f) ? (v + dv * dt) : v;           // refractory freeze
  float wn = w + dw * dt;
  ge = ge - (ge / tau_ge[i]) * dt;
  gi = gi - (gi / tau_gi[i]) * dt;

  const bool spk = vn > v_cut[i];
  vn = spk ? v_reset[i] : vn;
  wn = spk ? (wn + b_p[i]) : wn;
  const float rf = (spk ? t_refrac[i] : refr) - dt;

  out_v[i]  = vn;
  out_w[i]  = wn;
  out_ge[i] = ge;
  out_gi[i] = gi;
  out_rf[i] = rf;
  out_spk[i] = spk ? 1.0f : 0.0f;
}

// ---------------------------------------------------------------------------
extern "C" void kernel_launch(void* const* d_in, const int* in_sizes, int n_in,
                              void* d_out, int out_size, void* d_ws, size_t ws_size,
                              hipStream_t stream) {
  const float* voltage    = (const float*)d_in[0];
  const float* adapt      = (const float*)d_in[1];
  const float* ge_in      = (const float*)d_in[2];
  const float* gi_in      = (const float*)d_in[3];
  const float* stimulus   = (const float*)d_in[4];
  const float* refr       = (const float*)d_in[5];
  const float* g_L        = (const float*)d_in[6];
  const float* delta_T    = (const float*)d_in[7];
  const float* v_thresh   = (const float*)d_in[8];
  const float* v_rest     = (const float*)d_in[9];
  const float* Cm         = (const float*)d_in[10];
  const float* a_p        = (const float*)d_in[11];
  const float* b_p        = (const float*)d_in[12];
  const float* tau_w      = (const float*)d_in[13];
  const float* tau_ge     = (const float*)d_in[14];
  const float* tau_gi     = (const float*)d_in[15];
  const float* E_ge       = (const float*)d_in[16];
  const float* E_gi       = (const float*)d_in[17];
  const float* I_bias     = (const float*)d_in[18];
  const float* stim_scale = (const float*)d_in[19];
  const float* Q_ge       = (const float*)d_in[20];
  const float* Q_gi       = (const float*)d_in[21];
  const float* v_cut      = (const float*)d_in[22];
  const float* v_reset    = (const float*)d_in[23];
  const float* t_refrac   = (const float*)d_in[24];
  const unsigned char* spiked = (const unsigned char*)d_in[25];  // jax bool
  const unsigned char* is_exc = (const unsigned char*)d_in[26];  // jax bool
  const int*   edge_index = (const int*)d_in[27];                // (2, E) flat
  const float* dt_ptr     = (const float*)d_in[28];

  const int  n = in_sizes[0];
  const long E = (long)in_sizes[27] / 2;

  unsigned int*  cnt2 = (unsigned int*)d_ws;        // 2*n u32, interleaved
  unsigned char* code = (unsigned char*)(cnt2 + 2 * (long)n);  // n bytes

  float* out_v  = (float*)d_out;
  float* out_w  = out_v  + n;
  float* out_ge = out_w  + n;
  float* out_gi = out_ge + n;
  float* out_rf = out_gi + n;
  float* out_sp = out_rf + n;

  // Phase 0: spike code table + zero counters (deterministic, graph-safe)
  init_kernel<<<(n + TPB - 1) / TPB, TPB, 0, stream>>>(spiked, is_exc, code,
                                                       cnt2, n);

  // Phase 1: edge scatter (TDM + async-LDS staged edge stream)
  edge_accum_kernel<<<EDGE_BLOCKS, TPB, 0, stream>>>(
      edge_index, edge_index + E, E, code, cnt2);

  // Phase 2: neuron ODE update
  neuron_kernel<<<(n + TPB - 1) / TPB, TPB, 0, stream>>>(
      voltage, adapt, ge_in, gi_in, stimulus, refr,
      g_L, delta_T, v_thresh, v_rest, Cm, a_p, b_p, tau_w, tau_ge, tau_gi,
      E_ge, E_gi, I_bias, stim_scale, Q_ge, Q_gi, v_cut, v_reset, t_refrac,
      cnt2, dt_ptr,
      out_v, out_w, out_ge, out_gi, out_rf, out_sp, n);
}